// PQCPart_71459665871623
// MI455X (gfx1250) — compile-verified
//
#include <hip/hip_runtime.h>

// ---------------------------------------------------------------------------
// 18-qubit PQC state-vector simulator for MI455X (gfx1250, wave32).
//
//  * Hadamard wall computed analytically (one write pass).
//  * CNOT chain == Gray-map gather  g(i) = i ^ (i>>1), folded into the load
//    addressing of the following pass (zero-cost).
//  * 2 fused passes per layer: Rz diagonal + 4 register-resident Rx stages
//    + remaining Rx stages in LDS.  Global<->LDS movement uses CDNA5 async
//    LDS DMA (GLOBAL_LOAD_ASYNC_TO_LDS_B64 / GLOBAL_STORE_ASYNC_FROM_LDS_B64,
//    ASYNCcnt) when the toolchain exposes the builtins.
//  * Final <Z_j> = probs @ zvals GEMM via V_WMMA_F32_16X16X4_F32.
//
//  Workspace: double-buffered state, 2 * 32 * 2^18 * 8 B = 128 MiB
//  (fits MI455X 192 MB L2 -> passes run at L2, not HBM, bandwidth).
// ---------------------------------------------------------------------------

#define NQ        18
#define DIM       (1 << NQ)          // 262144
#define NBATCH    32
#define NLAYER    6
#define TPB       256
#define CHUNK_BITS 12
#define CHUNK     (1 << CHUNK_BITS)  // 4096 amplitudes = 32 KB LDS
#define APT       (CHUNK / TPB)      // 16 amps / thread

typedef float v8f __attribute__((ext_vector_type(8)));
typedef float v2f __attribute__((ext_vector_type(2)));

#if __has_builtin(__builtin_amdgcn_global_load_async_to_lds_b64) && \
    __has_builtin(__builtin_amdgcn_s_wait_asynccnt)
#define ASYNC_LDS_LOAD 1
#endif
#if defined(ASYNC_LDS_LOAD) && \
    __has_builtin(__builtin_amdgcn_global_store_async_from_lds_b64)
#define ASYNC_LDS_STORE 1
#endif

// Exact pointee type from the compiler diagnostic: GCC-style vector_size(8) int.
typedef int v2i_gcc __attribute__((vector_size(8)));
typedef __attribute__((address_space(1))) v2i_gcc* gp64_t;  // global (AS1)
typedef __attribute__((address_space(3))) v2i_gcc* lp64_t;  // LDS (AS3)

// -------------------------------------------------------------------- zero
__global__ void zero_out_kernel(float* __restrict__ out) {
  if (threadIdx.x < NBATCH) out[threadIdx.x] = 0.0f;
}

// -------------------------------------------------------------------- init
// After clip(s,0,1) the basis index is t = sum_q bit_q << (17-q) (big-endian).
// H^{x18}|t> : amp_i = 2^-9 * (-1)^popc(i & t), purely real.
__global__ __launch_bounds__(TPB)
void init_kernel(const int* __restrict__ sbits, float2* __restrict__ out) {
  const int b = blockIdx.y;
  unsigned t = 0u;
#pragma unroll
  for (int q = 0; q < NQ; ++q) {
    int v = sbits[b * NQ + q];
    v = v < 0 ? 0 : (v > 1 ? 1 : v);
    t |= (unsigned)v << (NQ - 1 - q);
  }
  const float a0 = 1.0f / 512.0f;  // 2^-9
  float2* o = out + (size_t)b * DIM;
  const int base = (blockIdx.x * TPB + threadIdx.x) * 8;
#pragma unroll
  for (int k = 0; k < 8; ++k) {
    const int i = base + k;
    const float s = (__popc((unsigned)i & t) & 1) ? -a0 : a0;
    o[i] = make_float2(s, 0.0f);
  }
}

// -------------------------------------------------------------- gate passes
// mode 0: local bits 0..11 of the index, chunk id = bits 12..17.
//         Full Rz diagonal; Rx on qubits 6..9 in registers, 10..17 in LDS.
//         Optionally folds previous layer's CNOT chain (Gray gather) on load.
// mode 1: local bits {0..5, 12..17}, chunk id = bits 6..11 (in place).
//         Rx on qubits 0..3 in registers (full bits 14..17), 4..5 in LDS.
// Thread t owns amplitudes a = t + u*256 (u = 0..15): LDS word address steps
// by 2 per lane -> all 64 banks hit exactly once, conflict-free; the owned
// u-bits are a-bits 8..11, which carry the register-resident butterflies.
__device__ __forceinline__ int map_index(int mode, int c, int a) {
  if (mode == 0) return (c << CHUNK_BITS) | a;
  return (a & 63) | (c << 6) | ((a >> 6) << 12);
}

__global__ __launch_bounds__(TPB)
void gate_pass_kernel(const float2* __restrict__ in, float2* __restrict__ out,
                      const float* __restrict__ thx,  // [NQ] this layer
                      const float* __restrict__ thz,  // [NQ] this layer
                      int mode, int applyGather) {
  __shared__ float2 amp[CHUNK];
  __shared__ float s_cx[NQ], s_sx[NQ], s_thz[NQ];
  __shared__ float s_sumz;

  const int b = blockIdx.y;
  const int c = blockIdx.x;
  const int tid = threadIdx.x;
  const float2* inB = in + (size_t)b * DIM;
  float2* outB = out + (size_t)b * DIM;

  if (tid < NQ) {
    const float tx = 0.5f * thx[tid];
    s_cx[tid] = __cosf(tx);
    s_sx[tid] = __sinf(tx);
    s_thz[tid] = thz[tid];
  }
  if (tid == 0) {
    float s = 0.0f;
    for (int q = 0; q < NQ; ++q) s += thz[q];
    s_sumz = s;
  }

  // ---- load (fold CNOT-chain Gray gather); async DMA global->LDS ----
#ifdef ASYNC_LDS_LOAD
#pragma unroll
  for (int r = 0; r < APT; ++r) {
    const int a = tid + r * TPB;
    const int i = map_index(mode, c, a);
    const int src = applyGather ? (i ^ (i >> 1)) : i;
    __builtin_amdgcn_global_load_async_to_lds_b64(
        (gp64_t)(inB + src), (lp64_t)&amp[a], 0, 0);
  }
  __builtin_amdgcn_s_wait_asynccnt(0);
#else
#pragma unroll
  for (int r = 0; r < APT; ++r) {
    const int a = tid + r * TPB;
    const int i = map_index(mode, c, a);
    const int src = applyGather ? (i ^ (i >> 1)) : i;
    __builtin_prefetch(&inB[src], 0, 1);  // global_prefetch_b8
    amp[a] = inB[src];
  }
#endif
  __syncthreads();

  // ---- register phase: Rz diagonal (mode 0) + 4 Rx stages on a-bits 8..11 ----
  {
    float2 r[16];
#pragma unroll
    for (int u = 0; u < 16; ++u) r[u] = amp[tid + (u << 8)];

    if (mode == 0) {
      // phase(i) = sum_{set bits q} thz[q] - 0.5 * sum thz ; u-bits are
      // i-bits 8..11 -> qubits 9,8,7,6; remaining bits folded into `base`.
      const int ibase = (c << CHUNK_BITS) | tid;  // u-bits are zero here
      float base = -0.5f * s_sumz;
#pragma unroll
      for (int q = 0; q < NQ; ++q)
        base += s_thz[q] * (float)((ibase >> (NQ - 1 - q)) & 1);
      const float t0 = s_thz[9], t1 = s_thz[8], t2 = s_thz[7], t3 = s_thz[6];
#pragma unroll
      for (int u = 0; u < 16; ++u) {
        float ph = base;
        if (u & 1) ph += t0;
        if (u & 2) ph += t1;
        if (u & 4) ph += t2;
        if (u & 8) ph += t3;
        float sp, cp;
        __sincosf(ph, &sp, &cp);
        const float2 v = r[u];
        r[u] = make_float2(v.x * cp - v.y * sp, v.x * sp + v.y * cp);
      }
    }

    // Rx butterflies on u-bit k: mode0 -> qubit 9-k ; mode1 -> qubit 3-k
#pragma unroll
    for (int k = 0; k < 4; ++k) {
      const int q = (mode == 0) ? (9 - k) : (3 - k);
      const float cq = s_cx[q], sq = s_sx[q];
#pragma unroll
      for (int u0 = 0; u0 < 16; ++u0) {
        if (u0 & (1 << k)) continue;
        const int u1 = u0 | (1 << k);
        const float2 x0 = r[u0];
        const float2 x1 = r[u1];
        // x0' = c*x0 - i s*x1 ; x1' = -i s*x0 + c*x1
        r[u0] = make_float2(cq * x0.x + sq * x1.y, cq * x0.y - sq * x1.x);
        r[u1] = make_float2(sq * x0.y + cq * x1.x, cq * x1.y - sq * x0.x);
      }
    }

#pragma unroll
    for (int u = 0; u < 16; ++u) amp[tid + (u << 8)] = r[u];
  }
  __syncthreads();

  // ---- remaining Rx stages in LDS ----
  // mode 0: m = 0..7 (full bit m,   qubit 17-m  -> qubits 17..10)
  // mode 1: m = 6..7 (full bit m+6, qubit 11-m  -> qubits 5,4)
  const int m_lo = (mode == 0) ? 0 : 6;
  for (int m = m_lo; m < 8; ++m) {
    const int fullbit = (mode == 0) ? m : (m + 6);
    const int q = NQ - 1 - fullbit;
    const float cq = s_cx[q], sq = s_sx[q];
    const int lowmask = (1 << m) - 1;
#pragma unroll
    for (int r = 0; r < APT / 2; ++r) {
      const int p = tid + r * TPB;  // pair id in [0, CHUNK/2)
      const int a0 = ((p & ~lowmask) << 1) | (p & lowmask);
      const int a1 = a0 | (1 << m);
      const float2 x0 = amp[a0];
      const float2 x1 = amp[a1];
      amp[a0] = make_float2(cq * x0.x + sq * x1.y, cq * x0.y - sq * x1.x);
      amp[a1] = make_float2(sq * x0.y + cq * x1.x, cq * x1.y - sq * x0.x);
    }
    __syncthreads();
  }

  // ---- store (linear in this pass's layout); async DMA LDS->global ----
#ifdef ASYNC_LDS_STORE
#pragma unroll
  for (int r = 0; r < APT; ++r) {
    const int a = tid + r * TPB;
    __builtin_amdgcn_global_store_async_from_lds_b64(
        (gp64_t)(outB + map_index(mode, c, a)), (lp64_t)&amp[a], 0, 0);
  }
  __builtin_amdgcn_s_wait_asynccnt(0);
#else
#pragma unroll
  for (int r = 0; r < APT; ++r) {
    const int a = tid + r * TPB;
    outB[map_index(mode, c, a)] = amp[a];
  }
#endif
}

// ------------------------------------------------------- expectation (WMMA)
// f[b] = sum_j coef[j] * sum_i |psi_b[i]|^2 * (1 - 2*((i>>j)&1))
// Real GEMM: probs[32 x 2^18] @ zvals[2^18 x 18]  ->  V_WMMA_F32_16X16X4_F32.
// Fragment layouts (ISA 7.12.2, wave32):
//   A 16x4 f32 : lane L, vgpr v -> A[M = L&15][K = 2*(L>>4) + v]
//   B 4x16 f32 : lane L, vgpr v -> B[K = 2*(L>>4) + v][N = L&15]
//   C 16x16 f32: lane L, vgpr v -> C[M = (L>>4)*8 + v][N = L&15]
#define KCHUNK 4096
#define KBLK   128
#define PROW   132   // LDS row stride (bank-conflict-free for column reads)

__global__ __launch_bounds__(TPB)
void expect_kernel(const float2* __restrict__ psi, const float* __restrict__ coef,
                   float* __restrict__ fout) {
  __shared__ float pblk[NBATCH][PROW];
  __shared__ float red[8][4][32][8];

  const int tid = threadIdx.x;
  const int lane = tid & 31;
  const int w = tid >> 5;       // wave id 0..7
  const int hl = lane >> 4;     // half-wave select
  const int ln = lane & 15;
  const int kbase = blockIdx.x * KCHUNK;

  const v8f vzero = {0.f, 0.f, 0.f, 0.f, 0.f, 0.f, 0.f, 0.f};
  v8f acc[2][2];
  acc[0][0] = vzero; acc[0][1] = vzero; acc[1][0] = vzero; acc[1][1] = vzero;

  for (int kb = 0; kb < KCHUNK; kb += KBLK) {
    // stage |psi|^2 for all 32 batches, folding last layer's CNOT gather
    for (int e = tid; e < NBATCH * KBLK; e += TPB) {
      const int b = e >> 7;        // KBLK == 128
      const int k = e & (KBLK - 1);
      const int i = kbase + kb + k;
      const int phys = i ^ (i >> 1);
      const float2 c2 = psi[(size_t)b * DIM + phys];
      pblk[b][k] = c2.x * c2.x + c2.y * c2.y;
      if (kb + KBLK < KCHUNK) {
        const int i2 = i + KBLK;
        __builtin_prefetch(&psi[(size_t)b * DIM + (i2 ^ (i2 >> 1))], 0, 1);
      }
    }
    __syncthreads();

    // each wave covers 16 k-values of this block -> 4 WMMA K-steps
#pragma unroll
    for (int s = 0; s < 4; ++s) {
      const int k0 = w * 16 + s * 4;
      const int kk = 2 * hl;           // this lane's K-slot base
      v2f a0, a1, b0, b1;
      a0.x = pblk[ln][k0 + kk];        a0.y = pblk[ln][k0 + kk + 1];
      a1.x = pblk[16 + ln][k0 + kk];   a1.y = pblk[16 + ln][k0 + kk + 1];
      const int iA = kbase + kb + k0 + kk;
      const int iB = iA + 1;
      const int j0 = ln;               // columns 0..15
      const int j1 = 16 + ln;          // columns 16..31 (>=18 zero-padded)
      b0.x = 1.0f - 2.0f * (float)((iA >> j0) & 1);
      b0.y = 1.0f - 2.0f * (float)((iB >> j0) & 1);
      b1.x = (j1 < NQ) ? (1.0f - 2.0f * (float)((iA >> j1) & 1)) : 0.0f;
      b1.y = (j1 < NQ) ? (1.0f - 2.0f * (float)((iB >> j1) & 1)) : 0.0f;

      acc[0][0] = __builtin_amdgcn_wmma_f32_16x16x4_f32(false, a0, false, b0,
                                                        (short)0, acc[0][0], false, false);
      acc[0][1] = __builtin_amdgcn_wmma_f32_16x16x4_f32(false, a0, false, b1,
                                                        (short)0, acc[0][1], false, false);
      acc[1][0] = __builtin_amdgcn_wmma_f32_16x16x4_f32(false, a1, false, b0,
                                                        (short)0, acc[1][0], false, false);
      acc[1][1] = __builtin_amdgcn_wmma_f32_16x16x4_f32(false, a1, false, b1,
                                                        (short)0, acc[1][1], false, false);
    }
    __syncthreads();
  }

  // cross-wave reduction of the four C tiles through LDS
#pragma unroll
  for (int mt = 0; mt < 2; ++mt)
#pragma unroll
    for (int nt = 0; nt < 2; ++nt) {
      const int combo = mt * 2 + nt;
#pragma unroll
      for (int v = 0; v < 8; ++v) red[w][combo][lane][v] = acc[mt][nt][v];
    }
  __syncthreads();

  if (tid < 32) {
    for (int combo = 0; combo < 4; ++combo) {
      const int mt = combo >> 1, nt = combo & 1;
#pragma unroll
      for (int v = 0; v < 8; ++v) {
        float sum = 0.0f;
#pragma unroll
        for (int ww = 0; ww < 8; ++ww) sum += red[ww][combo][tid][v];
        const int M = mt * 16 + (tid >> 4) * 8 + v;  // batch
        const int N = nt * 16 + (tid & 15);          // column j
        if (N < NQ) atomicAdd(&fout[M], sum * coef[N]);
      }
    }
  }
}

// ------------------------------------------------------------------- launch
extern "C" void kernel_launch(void* const* d_in, const int* in_sizes, int n_in,
                              void* d_out, int out_size, void* d_ws, size_t ws_size,
                              hipStream_t stream) {
  const int*   sbits = (const int*)d_in[0];     // [32,18] int32
  const float* coef  = (const float*)d_in[1];   // [18]
  const float* thx   = (const float*)d_in[2];   // [6,18]
  const float* thz   = (const float*)d_in[3];   // [6,18]
  float* fout = (float*)d_out;                  // [32]

  zero_out_kernel<<<1, 32, 0, stream>>>(fout);

  const size_t needed = 2ull * NBATCH * DIM * sizeof(float2);  // 128 MiB
  if (ws_size < needed) return;  // deterministic (zeroed) fallback

  float2* S0 = (float2*)d_ws;
  float2* S1 = S0 + (size_t)NBATCH * DIM;

  init_kernel<<<dim3(DIM / (TPB * 8), NBATCH), TPB, 0, stream>>>(sbits, S0);

  for (int l = 0; l < NLAYER; ++l) {
    const float2* inb = (l & 1) ? S1 : S0;
    float2* outb      = (l & 1) ? S0 : S1;
    gate_pass_kernel<<<dim3(DIM / CHUNK, NBATCH), TPB, 0, stream>>>(
        inb, outb, thx + l * NQ, thz + l * NQ, /*mode=*/0, /*gather=*/(l > 0) ? 1 : 0);
    gate_pass_kernel<<<dim3(DIM / CHUNK, NBATCH), TPB, 0, stream>>>(
        outb, outb, thx + l * NQ, thz + l * NQ, /*mode=*/1, /*gather=*/0);
  }

  // NLAYER even -> final state in S0; layer-5 CNOT chain folded into this read.
  expect_kernel<<<DIM / KCHUNK, TPB, 0, stream>>>(S0, coef, fout);
}